// EfficientSEA_Attention_Light_36515811951139
// MI455X (gfx1250) — compile-verified
//
#include <hip/hip_runtime.h>
#include <hip/hip_bf16.h>

// ---------------------------------------------------------------------------
// EfficientSEA axial attention, algebraically reduced:
//   qkv means commute with the 1x1 conv  -> only x.mean(W)/x.mean(H) needed
//   fusion output is separable: out(b,co,h,w) = A(b,co,h) + Bv(b,co,w)
//   gate mean over (H,W) = mean_h A + mean_w Bv
// GEMMs run on V_WMMA_F32_16X16X4_F32 (fp32 WMMA, wave32).
// MI455X-specific data movement:
//   - x (134 MB) fits in the 192 MB global L2: pass 1 warms it, pass 4 re-reads
//     it from L2; output stores are non-temporal so they don't evict x.
//   - pass 1 uses GLOBAL_LOAD_ASYNC_TO_LDS_B128 (ASYNCcnt) to fill LDS
//     without staging through VGPRs.
// ---------------------------------------------------------------------------

typedef __attribute__((ext_vector_type(2))) float v2f;
typedef __attribute__((ext_vector_type(4))) float v4f;
typedef __attribute__((ext_vector_type(8))) float v8f;

#define BATCH   32
#define DIM     256
#define HW      64          // H == W == 64
#define NHKD    16
#define QKV_CH  48          // 3 * NHKD
#define SCALE_F 0.35355339059327373f   // 8^-0.5

// ------------------------------ workspace layout (floats) ------------------
#define WS_XH   0                       // B*256*64
#define WS_XW   (BATCH*DIM*HW)          // B*256*64
#define WS_AG   (2*BATCH*DIM*HW)        // B*2*256*64  (axis0 = A(h), axis1 = Bv(w))
#define WS_GG   (WS_AG + 2*BATCH*DIM*HW)// B*256

// ===========================================================================
// Kernel 1: per-(b,c) plane row/col means of x   (bandwidth pass #1)
// ===========================================================================
__global__ __launch_bounds__(256) void sea_reduce_kernel(
    const float* __restrict__ x, float* __restrict__ xh, float* __restrict__ xw)
{
    __shared__ float tile[64 * 65];
    const int plane = blockIdx.x;                      // b*256 + c
    const int t = threadIdx.x;
    const float* xbase = x + (size_t)plane * 4096;

#if defined(__gfx1250__)
    // Async DMA: global -> LDS, 16B per lane per issue, no VGPR staging.
    // Low 32 bits of a generic pointer to LDS are the wave's LDS byte offset.
#pragma unroll
    for (int j = 0; j < 4; ++j) {
        int idx4 = t + j * 256;
        int e = idx4 * 4, h = e >> 6, w = e & 63;      // 16B chunk stays in one row
        unsigned ldsoff = (unsigned)(uintptr_t)(&tile[h * 65 + w]);
        const float* gptr = xbase + e;
        asm volatile("global_load_async_to_lds_b128 %0, %1, off"
                     :: "v"(ldsoff), "v"(gptr) : "memory");
    }
    asm volatile("s_wait_asynccnt 0x0" ::: "memory");
#else
    const v4f* xp = (const v4f*)xbase;
#pragma unroll
    for (int j = 0; j < 4; ++j) {
        int idx4 = t + j * 256;
        v4f v = xp[idx4];
        int e = idx4 * 4, h = e >> 6, w = e & 63;
        float* dst = &tile[h * 65 + w];
        dst[0] = v.x; dst[1] = v.y; dst[2] = v.z; dst[3] = v.w;
    }
#endif
    __syncthreads();
    if (t < 64) {                                      // mean over W -> xh(b,c,h)
        float s = 0.f;
#pragma unroll 8
        for (int w = 0; w < 64; ++w) s += tile[t * 65 + w];
        xh[(size_t)plane * 64 + t] = s * (1.0f / 64.0f);
    } else if (t < 128) {                              // mean over H -> xw(b,c,w)
        int w = t - 64;
        float s = 0.f;
#pragma unroll 8
        for (int h = 0; h < 64; ++h) s += tile[h * 65 + w];
        xw[(size_t)plane * 64 + w] = s * (1.0f / 64.0f);
    }
}

// ===========================================================================
// Kernel 2: per-(b,axis) qkv projection (WMMA) + axial attention + fusion GEMM
//           (WMMA). 384 threads = 12 wave32s.
// ===========================================================================
__global__ __launch_bounds__(384) void sea_mid_kernel(
    const float* __restrict__ xh, const float* __restrict__ xw,
    const float* __restrict__ w_qkv, const float* __restrict__ qkv_scale,
    const float* __restrict__ qkv_shift,
    const float* __restrict__ pos_h, const float* __restrict__ pos_w,
    const float* __restrict__ w_fusion, const float* __restrict__ fusion_scale,
    const float* __restrict__ fusion_shift,
    float* __restrict__ Ag)
{
    const int bb = blockIdx.x >> 1;
    const int axis = blockIdx.x & 1;
    const float* __restrict__ xm  = (axis == 0 ? xh : xw) + (size_t)bb * DIM * HW;
    const float* __restrict__ pos = (axis == 0 ? pos_h : pos_w);

    __shared__ float qs[NHKD * 65];
    __shared__ float ks_[NHKD * 65];
    __shared__ float vs[NHKD * 65];
    __shared__ float sc[2 * 64 * 65];   // per-head scores
    __shared__ float ao[NHKD * 65];     // axial attention output

    const int tid = threadIdx.x;
    const int lane = tid & 31;          // wave32
    const int wid = tid >> 5;           // 0..11
    const int lm = lane & 15;
    const int khalf = lane >> 4;        // 0: K {0,1}, 1: K {2,3}

    // ---- qkv projection: (48 x 256) @ (256 x 64), 3x4 tiles, 1 tile/wave ----
    {
        const int mt = wid >> 2, nt = wid & 3;
        const int row = mt * 16 + lm;   // output channel (A: M = lane%16)
        const int col = nt * 16 + lm;   // L             (B/D: N = lane%16)
        v8f acc = {};
        for (int k = 0; k < DIM; k += 4) {
            const int k0 = k + khalf * 2;
            v2f a, b;
            a.x = w_qkv[row * DIM + k0];
            a.y = w_qkv[row * DIM + k0 + 1];
            b.x = xm[k0 * HW + col];
            b.y = xm[(k0 + 1) * HW + col];
            acc = __builtin_amdgcn_wmma_f32_16x16x4_f32(
                false, a, false, b, (short)0, acc, false, false);
        }
#pragma unroll
        for (int r = 0; r < 8; ++r) {
            const int m = mt * 16 + r + khalf * 8;     // D: VGPR r -> M=r / r+8
            float val = acc[r] * qkv_scale[m] + qkv_shift[m];
            if (m < NHKD)            qs[m * 65 + col]        = val + pos[m * HW + col];
            else if (m < 2 * NHKD)   ks_[(m - 16) * 65 + col] = val + pos[(m - 16) * HW + col];
            else                     vs[(m - 32) * 65 + col]  = val;
        }
    }
    __syncthreads();

    // ---- scores: s[h][i][j] = SCALE * sum_d q[d,i] k[d,j] ----
    for (int idx = tid; idx < 2 * 64 * 64; idx += 384) {
        const int hd = idx >> 12, rem = idx & 4095, i = rem >> 6, j = rem & 63;
        const int cb = hd * 8;
        float s = 0.f;
#pragma unroll
        for (int d = 0; d < 8; ++d) s += qs[(cb + d) * 65 + i] * ks_[(cb + d) * 65 + j];
        sc[(hd * 64 + i) * 65 + j] = s * SCALE_F;
    }
    __syncthreads();

    // ---- softmax over j (128 rows) ----
    if (tid < 128) {
        float* rowp = &sc[tid * 65];
        float mx = rowp[0];
#pragma unroll 8
        for (int j = 1; j < 64; ++j) mx = fmaxf(mx, rowp[j]);
        float sum = 0.f;
#pragma unroll 8
        for (int j = 0; j < 64; ++j) { float e = __expf(rowp[j] - mx); rowp[j] = e; sum += e; }
        const float inv = 1.0f / sum;
#pragma unroll 8
        for (int j = 0; j < 64; ++j) rowp[j] *= inv;
    }
    __syncthreads();

    // ---- out: ao[ch][i] = sum_j attn[i,j] * v[ch,j] ----
    for (int idx = tid; idx < NHKD * 64; idx += 384) {
        const int ch = idx >> 6, i = idx & 63, hd = ch >> 3;
        const float* arow = &sc[(hd * 64 + i) * 65];
        const float* vrow = &vs[ch * 65];
        float s = 0.f;
#pragma unroll 8
        for (int j = 0; j < 64; ++j) s += arow[j] * vrow[j];
        ao[ch * 65 + i] = s;
    }
    __syncthreads();

    // ---- fusion GEMM: (256 x 16) @ (16 x 64) -> A (axis0) / Bv (axis1) ----
    float* __restrict__ Aout = Ag + (size_t)blockIdx.x * DIM * HW;
    for (int t = wid; t < 64; t += 12) {               // wave-uniform -> EXEC all 1s
        const int mt = t >> 2, nt = t & 3;
        const int row = mt * 16 + lm;                  // co
        const int col = nt * 16 + lm;                  // L
        v8f acc = {};
#pragma unroll
        for (int k = 0; k < NHKD; k += 4) {
            const int k0 = k + khalf * 2;
            v2f a, b;
            a.x = w_fusion[row * 32 + axis * 16 + k0];
            a.y = w_fusion[row * 32 + axis * 16 + k0 + 1];
            b.x = ao[k0 * 65 + col];
            b.y = ao[(k0 + 1) * 65 + col];
            acc = __builtin_amdgcn_wmma_f32_16x16x4_f32(
                false, a, false, b, (short)0, acc, false, false);
        }
#pragma unroll
        for (int r = 0; r < 8; ++r) {
            const int co = mt * 16 + r + khalf * 8;
            float val = acc[r] * fusion_scale[co] + (axis == 0 ? fusion_shift[co] : 0.0f);
            Aout[co * HW + col] = val;
        }
    }
}

// ===========================================================================
// Kernel 3: SE gate per batch: mean(A)+mean(Bv) -> MLP -> sigmoid, *0.1
// ===========================================================================
__global__ __launch_bounds__(256) void sea_gate_kernel(
    const float* __restrict__ Ag,
    const float* __restrict__ w_g1, const float* __restrict__ g1s, const float* __restrict__ g1b,
    const float* __restrict__ w_g2, const float* __restrict__ g2s, const float* __restrict__ g2b,
    float* __restrict__ gg)
{
    __shared__ float m[DIM];
    __shared__ float y1[64];
    const int b = blockIdx.x, t = threadIdx.x;
    const float* Ab = Ag + (size_t)(b * 2) * DIM * HW;
    float s = 0.f;
#pragma unroll 8
    for (int L = 0; L < 64; ++L)
        s += Ab[t * 64 + L] + Ab[DIM * HW + t * 64 + L];
    m[t] = s * (1.0f / 64.0f);     // mean_h A + mean_w Bv
    __syncthreads();
    if (t < 64) {
        float a = 0.f;
        for (int c = 0; c < DIM; ++c) a += w_g1[t * DIM + c] * m[c];
        a = a * g1s[t] + g1b[t];
        y1[t] = a / (1.0f + __expf(-a));     // silu
    }
    __syncthreads();
    float z = 0.f;
#pragma unroll 8
    for (int o = 0; o < 64; ++o) z += w_g2[t * 64 + o] * y1[o];
    z = z * g2s[t] + g2b[t];
    gg[(size_t)b * DIM + t] = 0.1f / (1.0f + __expf(-z));   // sigmoid * 0.1
}

// ===========================================================================
// Kernel 4: out = x + (A[h] + Bv[w]) * g    (bandwidth pass #2)
// x reads hit L2 (warmed by pass 1; 134 MB < 192 MB); out stores are
// non-temporal (TH_NT) so the streaming writes don't evict x mid-pass.
// ===========================================================================
__global__ __launch_bounds__(256) void sea_final_kernel(
    const float* __restrict__ x, const float* __restrict__ Ag,
    const float* __restrict__ gg, float* __restrict__ out)
{
    __shared__ float Ah[64], Bw[64];
    const int plane = blockIdx.x;              // b*256 + c
    const int b = plane >> 8, c = plane & 255;
    const int t = threadIdx.x;
    if (t < 64)        Ah[t]      = Ag[((size_t)(b * 2) * DIM + c) * 64 + t];
    else if (t < 128)  Bw[t - 64] = Ag[((size_t)(b * 2 + 1) * DIM + c) * 64 + (t - 64)];
    const float ga = gg[plane];
    __syncthreads();
    const v4f* xp = (const v4f*)(x + (size_t)plane * 4096);
    v4f* op = (v4f*)(out + (size_t)plane * 4096);
#pragma unroll
    for (int j = 0; j < 4; ++j) {
        int idx4 = t + j * 256;
        v4f v = xp[idx4];                              // RT load: hits L2
        int e = idx4 * 4, h = e >> 6, w = e & 63;
        const float a = Ah[h];
        v4f r;
        r.x = v.x + (a + Bw[w + 0]) * ga;
        r.y = v.y + (a + Bw[w + 1]) * ga;
        r.z = v.z + (a + Bw[w + 2]) * ga;
        r.w = v.w + (a + Bw[w + 3]) * ga;
        __builtin_nontemporal_store(r, &op[idx4]);     // TH_NT: keep x in L2
    }
}

// ===========================================================================
extern "C" void kernel_launch(void* const* d_in, const int* in_sizes, int n_in,
                              void* d_out, int out_size, void* d_ws, size_t ws_size,
                              hipStream_t stream) {
    (void)in_sizes; (void)n_in; (void)out_size; (void)ws_size;
    const float* x            = (const float*)d_in[0];
    const float* w_qkv        = (const float*)d_in[1];
    const float* qkv_scale    = (const float*)d_in[2];
    const float* qkv_shift    = (const float*)d_in[3];
    const float* pos_h        = (const float*)d_in[4];
    const float* pos_w        = (const float*)d_in[5];
    const float* w_fusion     = (const float*)d_in[6];
    const float* fusion_scale = (const float*)d_in[7];
    const float* fusion_shift = (const float*)d_in[8];
    const float* w_g1         = (const float*)d_in[9];
    const float* g1_scale     = (const float*)d_in[10];
    const float* g1_shift     = (const float*)d_in[11];
    const float* w_g2         = (const float*)d_in[12];
    const float* g2_scale     = (const float*)d_in[13];
    const float* g2_shift     = (const float*)d_in[14];
    float* out = (float*)d_out;

    float* ws = (float*)d_ws;
    float* xh = ws + WS_XH;
    float* xw = ws + WS_XW;
    float* Ag = ws + WS_AG;
    float* gg = ws + WS_GG;

    sea_reduce_kernel<<<BATCH * DIM, 256, 0, stream>>>(x, xh, xw);
    sea_mid_kernel<<<BATCH * 2, 384, 0, stream>>>(
        xh, xw, w_qkv, qkv_scale, qkv_shift, pos_h, pos_w,
        w_fusion, fusion_scale, fusion_shift, Ag);
    sea_gate_kernel<<<BATCH, 256, 0, stream>>>(
        Ag, w_g1, g1_scale, g1_shift, w_g2, g2_scale, g2_shift, gg);
    sea_final_kernel<<<BATCH * DIM, 256, 0, stream>>>(x, Ag, gg, out);
}